// TimeWarping_77163382440321
// MI455X (gfx1250) — compile-verified
//
#include <hip/hip_runtime.h>
#include <stdint.h>

// Problem constants (fixed by the reference)
#define LEADS 512
#define TLEN 65536
#define TILES_PER_LEAD 16
#define TILE_T (TLEN / TILES_PER_LEAD) // 4096
#define PSTRIDE 16                     // floats per lead in workspace

#define HKN 13107.0f                     // (T-1)/5, exact in fp32
#define INV_HKN (1.0f / 13107.0f)
#define H2_6 ((13107.0f * 13107.0f) / 6.0f)
#define H_6 (13107.0f / 6.0f)

#define USE_ASYNC_LDS 1

typedef __attribute__((ext_vector_type(2))) float v2f;
typedef __attribute__((ext_vector_type(8))) float v8f;

__device__ __forceinline__ float frcp_fast(float v) {
#if __has_builtin(__builtin_amdgcn_rcpf)
  return __builtin_amdgcn_rcpf(v);
#else
  return 1.0f / v;
#endif
}

__device__ __forceinline__ float clampT(float v) {
  v = v < 0.0f ? 0.0f : v;
  return v > 65535.0f ? 65535.0f : v;
}

// Reference-form xp evaluation at integer j (matches reference seg/a/c formula).
__device__ __forceinline__ float xp_eval(int j, const float* sp, float scale) {
  float tf = (float)j;
  int s = (int)(tf * INV_HKN);
  s = s > 4 ? 4 : s;
  float x0 = (float)s * HKN;                 // exact
  float a = (x0 + HKN - tf) * INV_HKN;       // (xs[s+1]-t)/h
  float c = (tf - x0) * INV_HKN;             // (t-xs[s])/h
  float y0 = sp[s], y1 = sp[s + 1];
  float m0 = sp[6 + s], m1 = sp[7 + s];
  float tw = a * y0 + c * y1 + ((a * a * a - a) * m0 + (c * c * c - c) * m1) * H2_6;
  return clampT(scale * tw);
}

// Value + derivative at continuous jf within fixed segment s (unclamped, for Newton).
__device__ __forceinline__ void xp_eval_grad(float tf, int s, const float* sp,
                                             float scale, float& v, float& dv) {
  float x0 = (float)s * HKN;
  float a = (x0 + HKN - tf) * INV_HKN;
  float c = (tf - x0) * INV_HKN;
  float y0 = sp[s], y1 = sp[s + 1];
  float m0 = sp[6 + s], m1 = sp[7 + s];
  float tw = a * y0 + c * y1 + ((a * a * a - a) * m0 + (c * c * c - c) * m1) * H2_6;
  float dtw = (y1 - y0) * INV_HKN +
              ((3.0f * c * c - 1.0f) * m1 - (3.0f * a * a - 1.0f) * m0) * H_6;
  v = scale * tw;
  dv = scale * dtw;
}

// Exact inverse of the 6x6 not-a-knot matrix, restricted to columns 1..4
// (b[0]==b[5]==0), derived in closed form. Row-major [knot][b-index-1].
__device__ static const float AINV[6][4] = {
    {17.0f / 45.0f, -4.0f / 15.0f, 1.0f / 15.0f, -1.0f / 90.0f},
    {1.0f / 6.0f, 0.0f, 0.0f, 0.0f},
    {-2.0f / 45.0f, 4.0f / 15.0f, -1.0f / 15.0f, 1.0f / 90.0f},
    {1.0f / 90.0f, -1.0f / 15.0f, 4.0f / 15.0f, -2.0f / 45.0f},
    {0.0f, 0.0f, 0.0f, 1.0f / 6.0f},
    {-1.0f / 90.0f, 1.0f / 15.0f, -4.0f / 15.0f, 17.0f / 45.0f}};

// ---------------- Phase 1: per-lead spline setup --------------------------------
// The reference's only matrix op (jnp.linalg.solve with b of shape (6,512)) is
// lowered to V_WMMA_F32_16X16X4_F32: M(6x16-lead-tile) = Ainv'(6x4) @ b'(4x16).
// One wave (32 lanes, EXEC all ones) per 16-lead tile; 32 tiles.
__global__ __launch_bounds__(32) void TimeWarping_77163382440321_setup(
    const float* __restrict__ W, float* __restrict__ P) {
  const int lane = threadIdx.x;        // 0..31
  const int tile = blockIdx.x;         // 0..31
  const int n = lane & 15;             // column (lead within tile)
  const int lead = tile * 16 + n;
  const bool hiHalf = lane >= 16;

  // Every lane loads the warps of its column's lead (hi half needs b3/b4).
  float ys[6];
#pragma unroll
  for (int k = 0; k < 6; ++k) ys[k] = ((float)k * HKN) * W[k * LEADS + lead];

  const float c6h2 = 6.0f / (13107.0f * 13107.0f);
  float b1 = c6h2 * (ys[2] - 2.0f * ys[1] + ys[0]);
  float b2 = c6h2 * (ys[3] - 2.0f * ys[2] + ys[1]);
  float b3 = c6h2 * (ys[4] - 2.0f * ys[3] + ys[2]);
  float b4 = c6h2 * (ys[5] - 2.0f * ys[4] + ys[3]);

  float Mv[6];
#if __has_builtin(__builtin_amdgcn_wmma_f32_16x16x4_f32)
  // B operand (4x16 f32): lanes 0-15 hold K=0,1 ; lanes 16-31 hold K=2,3.
  v2f Bop;
  Bop.x = hiHalf ? b3 : b1;
  Bop.y = hiHalf ? b4 : b2;
  // A operand (16x4 f32): lane m holds row m; VGPR0=K0(lo)/K2(hi), VGPR1=K1/K3.
  v2f Aop;
  Aop.x = (n < 6) ? (hiHalf ? AINV[n][2] : AINV[n][0]) : 0.0f;
  Aop.y = (n < 6) ? (hiHalf ? AINV[n][3] : AINV[n][1]) : 0.0f;
  v8f acc = {};
  acc = __builtin_amdgcn_wmma_f32_16x16x4_f32(
      /*neg_a=*/false, Aop, /*neg_b=*/false, Bop,
      /*c_mod=*/(short)0, acc, /*reuse_a=*/false, /*reuse_b=*/false);
  // D (16x16 f32): lane n (0-15) holds rows 0..7 in acc[0..7]; rows 0..5 = M.
#pragma unroll
  for (int k = 0; k < 6; ++k) Mv[k] = acc[k];
#else
  // Closed-form fallback (same math as the hardcoded inverse).
  float x1 = b1 * (1.0f / 6.0f);
  float x4 = b4 * (1.0f / 6.0f);
  float r2 = b2 - x1;
  float r3 = b3 - x4;
  float x2 = (4.0f * r2 - r3) * (1.0f / 15.0f);
  float x3 = (4.0f * r3 - r2) * (1.0f / 15.0f);
  Mv[0] = 2.0f * x1 - x2; Mv[1] = x1; Mv[2] = x2;
  Mv[3] = x3; Mv[4] = x4; Mv[5] = 2.0f * x4 - x3;
#endif

  if (!hiHalf) {  // lanes 0-15: one lead each -> epilogue + store
    float scale = 65535.0f / ys[5];  // tw[:,-1] == ys[5] exactly in the reference
    bool mono = (scale > 0.0f) && (scale < 3.0e38f) && (scale == scale);
    for (int s = 0; s < 5 && mono; ++s) {
      float y0 = ys[s], y1 = ys[s + 1], m0 = Mv[s], m1 = Mv[s + 1];
      auto g = [&](float c) {
        float a = 1.0f - c;
        return (y1 - y0) * INV_HKN +
               ((3.0f * c * c - 1.0f) * m1 - (3.0f * a * a - 1.0f) * m0) * H_6;
      };
      mono = (g(0.0f) >= 0.0f) && (g(1.0f) >= 0.0f);
      float dm = m0 - m1;
      if (mono && dm != 0.0f) {
        float cs = m0 / dm;  // vertex of the (quadratic) derivative
        if (cs > 0.0f && cs < 1.0f) mono = (g(cs) >= 0.0f);
      }
    }
    float* Pl = P + (size_t)lead * PSTRIDE;
#pragma unroll
    for (int k = 0; k < 6; ++k) { Pl[k] = ys[k]; Pl[6 + k] = Mv[k]; }
    Pl[12] = scale;
    Pl[13] = mono ? 1.0f : 0.0f;
    Pl[14] = 0.0f; Pl[15] = 0.0f;
  }
}

// ---------------- Phase 2: invert warp + resample ----------------
__global__ __launch_bounds__(256) void TimeWarping_77163382440321_interp(
    const float* __restrict__ x, const float* __restrict__ P, float* __restrict__ out) {
  __shared__ float sP[PSTRIDE];
  __shared__ float sTab[1024];  // xp at dyadic binary-search nodes (multiples of 64)

  const int tid = threadIdx.x;
  const int lead = blockIdx.x >> 4;
  const int tile = blockIdx.x & (TILES_PER_LEAD - 1);
  const float* Pl = P + (size_t)lead * PSTRIDE;

  if (tid < PSTRIDE) {
#if USE_ASYNC_LDS
    // CDNA5 async copy global->LDS (tracked by ASYNCcnt), one b32 per lane.
    unsigned ldsOff = (unsigned)(unsigned long long)&sP[tid];  // low 32b = LDS offset
    unsigned vOff = (unsigned)tid * 4u;
    asm volatile("global_load_async_to_lds_b32 %0, %1, %2"
                 :: "v"(ldsOff), "v"(vOff), "s"(Pl) : "memory");
#else
    sP[tid] = Pl[tid];
#endif
  }
#if USE_ASYNC_LDS
#if __has_builtin(__builtin_amdgcn_s_wait_asynccnt)
  __builtin_amdgcn_s_wait_asynccnt(0);
#else
  asm volatile("s_wait_asynccnt 0" ::: "memory");
#endif
#endif
  __syncthreads();

  const float scale = sP[12];
  const bool mono = (sP[13] != 0.0f);  // uniform per block (same lead)

  float K[6];
#pragma unroll
  for (int s = 0; s < 6; ++s) K[s] = clampT(scale * sP[s]);

  if (!mono) {
    // First 10 binary-search levels only ever probe j = g*64, g in [1,1023].
    for (int g = tid + 1; g < 1024; g += 256) sTab[g] = xp_eval(g << 6, sP, scale);
    __syncthreads();
  }

  const float* xrow = x + ((size_t)lead << 16);
  float* orow = out + ((size_t)lead << 16);
  const int t0 = tile * TILE_T;

  for (int k = 0; k < TILE_T / 256; ++k) {
    const int t = t0 + (k << 8) + tid;
    const float tf = (float)t;
    __builtin_prefetch(&xrow[t], 0, 1);  // idx(t) ~ t; warm the gather stream

    int jg;
    if (mono) {
      // segment of t among knot images, linear init, 3 Newton steps
      int s = (tf >= K[1]) + (tf >= K[2]) + (tf >= K[3]) + (tf >= K[4]);
      float den = K[s + 1] - K[s];
      float u01 = den > 0.0f ? (tf - K[s]) * frcp_fast(den) : 0.0f;
      float jf = ((float)s + u01) * HKN;
      const float jlo = (float)s * HKN;
      const float jhi = jlo + HKN;
#pragma unroll
      for (int it = 0; it < 3; ++it) {
        float v, dv;
        xp_eval_grad(jf, s, sP, scale, v, dv);
        jf -= (v - tf) * frcp_fast(dv);
        jf = jf < jlo ? jlo : (jf > jhi ? jhi : jf);
      }
      jg = (int)jf;
      if (jg > 65535) jg = 65535;
      // exact fix-up: last j with xp[j] <= t (== searchsorted for monotone xp)
      float xj = xp_eval(jg, sP, scale);
      while (jg > 0 && xj > tf) { --jg; xj = xp_eval(jg, sP, scale); }
      while (jg < 65535) {
        float xn = xp_eval(jg + 1, sP, scale);
        if (xn <= tf) { ++jg; xj = xn; } else break;
      }
    } else {
      // literal searchsorted(side='right') binary search over virtual xp[0..65535]
      int lo = 0, hi = 65536;
#pragma unroll
      for (int lvl = 0; lvl < 10; ++lvl) {  // mids provably multiples of 64 -> LDS table
        int mid = (lo + hi) >> 1;
        float v = sTab[mid >> 6];
        if (v <= tf) lo = mid + 1; else hi = mid;
      }
      while (lo < hi) {
        int mid = (lo + hi) >> 1;
        float v = xp_eval(mid, sP, scale);
        if (v <= tf) lo = mid + 1; else hi = mid;
      }
      jg = lo - 1;
      if (jg < 0) jg = 0;
    }

    const int idx = jg > 65534 ? 65534 : jg;
    const float X0 = xp_eval(idx, sP, scale);
    const float X1 = xp_eval(idx + 1, sP, scale);
    const float dx = X1 - X0;
    const float w = dx > 0.0f ? (tf - X0) / dx : 0.0f;
    const float f0 = xrow[idx];
    const float f1 = xrow[idx + 1];
    orow[t] = f0 + w * (f1 - f0);
  }
}

extern "C" void kernel_launch(void* const* d_in, const int* in_sizes, int n_in,
                              void* d_out, int out_size, void* d_ws, size_t ws_size,
                              hipStream_t stream) {
  (void)in_sizes; (void)n_in; (void)out_size; (void)ws_size;
  const float* x = (const float*)d_in[0];      // (512, 65536) fp32
  const float* warps = (const float*)d_in[1];  // (6, 512) fp32
  float* out = (float*)d_out;                  // (512, 65536) fp32
  float* P = (float*)d_ws;                     // 512*16 floats = 32 KB scratch

  TimeWarping_77163382440321_setup<<<32, 32, 0, stream>>>(warps, P);
  TimeWarping_77163382440321_interp<<<LEADS * TILES_PER_LEAD, 256, 0, stream>>>(x, P, out);
}